// LiteMLA_35192962023577
// MI455X (gfx1250) — compile-verified
//
#include <hip/hip_runtime.h>

typedef __attribute__((ext_vector_type(2))) float v2f;
typedef __attribute__((ext_vector_type(4))) float v4f;
typedef __attribute__((ext_vector_type(8))) float v8f;

#define HW      4096   // 64*64
#define CIN     256
#define C3      768
#define BATCH   8
#define ATT_C   512    // 64 heads * dim 8
#define KCHUNK  32
#define BSTRIDE 132    // floats per LDS row-pair (64*2 data + 4 pad, 16B-aligned)

// ---------------------------------------------------------------------------
// Shared GEMM core: one 256-thread block owns a 128(M) x 64(N) tile.
// B tile staged in LDS in *fragment-paired* layout: element (k,n) lives at
// Bs[(k>>1)*BSTRIDE + 2n + (k&1)], so a WMMA B fragment (rows kr,kr+1 at
// column n) is one aligned ds_load_b64 straight into the operand pair.
// Each wave runs V_WMMA_F32_16X16X4_F32 on a 16x64 sub-tile.
// ---------------------------------------------------------------------------
template <int K>
__device__ __forceinline__ void gemm_core(
    const float* __restrict__ A,    // weights, M x K (row-major)
    const float* __restrict__ Bp,   // input,   K x HW (row-major, this batch)
    float* __restrict__ Bs,         // LDS, 16 * BSTRIDE floats
    int m0, int n0, int l, int hi, int tid,
    v8f& acc0, v8f& acc1, v8f& acc2, v8f& acc3)
{
    const float* arow = A + (size_t)(m0 + l) * K + 2 * hi;   // A frag: M=l, K base 2*hi

    for (int k0 = 0; k0 < K; k0 += KCHUNK) {
        __syncthreads();   // previous chunk's LDS reads complete
        #pragma unroll
        for (int i = 0; i < 2; ++i) {
            const int idx = tid + i * 256;        // 0..511 2x2 micro-tiles
            const int kp  = idx >> 5;             // row-pair 0..15
            const int np  = idx & 31;             // col-pair 0..31
            const float* g = Bp + (size_t)(k0 + 2 * kp) * HW + n0 + 2 * np;
            const v2f r0 = *(const v2f*)g;        // row 2kp,   cols 2np..2np+1
            const v2f r1 = *(const v2f*)(g + HW); // row 2kp+1, cols 2np..2np+1
            v4f q; q.x = r0.x; q.y = r1.x; q.z = r0.y; q.w = r1.y;
            *(v4f*)&Bs[kp * BSTRIDE + 4 * np] = q;   // 16B-aligned b128 store
        }
        __syncthreads();

        #pragma unroll
        for (int kk = 0; kk < KCHUNK; kk += 4) {
            v2f a; a.x = arow[k0 + kk]; a.y = arow[k0 + kk + 1];
            const int rp = (kk >> 1) + hi;        // (kr>>1), kr = kk + 2*hi
            const float* brow = &Bs[rp * BSTRIDE + 2 * l];
            const v2f b0 = *(const v2f*)(brow + 0);
            const v2f b1 = *(const v2f*)(brow + 32);
            const v2f b2 = *(const v2f*)(brow + 64);
            const v2f b3 = *(const v2f*)(brow + 96);
            acc0 = __builtin_amdgcn_wmma_f32_16x16x4_f32(false, a, false, b0, (short)0, acc0, false, false);
            acc1 = __builtin_amdgcn_wmma_f32_16x16x4_f32(false, a, false, b1, (short)0, acc1, false, false);
            acc2 = __builtin_amdgcn_wmma_f32_16x16x4_f32(false, a, false, b2, (short)0, acc2, false, false);
            acc3 = __builtin_amdgcn_wmma_f32_16x16x4_f32(false, a, false, b3, (short)0, acc3, false, false);
        }
    }
}

// ---------------------------------------------------------------------------
// qkv = W_qkv (768x256) @ x   per batch image
// ---------------------------------------------------------------------------
__global__ __launch_bounds__(256) void gemm_qkv_kernel(
    const float* __restrict__ W,   // 768 x 256
    const float* __restrict__ X,   // B x 256 x 4096
    float* __restrict__ Y)         // B x 768 x 4096
{
    __shared__ float Bs[16 * BSTRIDE];
    const int tid  = threadIdx.x;
    const int lane = tid & 31;
    const int wave = tid >> 5;
    const int l    = lane & 15;
    const int hi   = lane >> 4;
    const int n0   = blockIdx.x * 64;
    const int m0   = (blockIdx.y * 8 + wave) * 16;
    const int b    = blockIdx.z;

    const float* Xb = X + (size_t)b * CIN * HW;
    float*       Yb = Y + (size_t)b * C3  * HW;

    v8f acc0 = {}, acc1 = {}, acc2 = {}, acc3 = {};
    gemm_core<CIN>(W, Xb, Bs, m0, n0, l, hi, tid, acc0, acc1, acc2, acc3);

    float* yp = Yb + (size_t)(m0 + 8 * hi) * HW + n0 + l;    // D row = m0 + r + 8*hi
    #pragma unroll
    for (int r = 0; r < 8; ++r) {
        yp[(size_t)r * HW + 0]  = acc0[r];
        yp[(size_t)r * HW + 16] = acc1[r];
        yp[(size_t)r * HW + 32] = acc2[r];
        yp[(size_t)r * HW + 48] = acc3[r];
    }
}

// ---------------------------------------------------------------------------
// Fused depthwise 5x5 (pad 2) + grouped 1x1 (96 groups, 8->8).
// ---------------------------------------------------------------------------
__global__ __launch_bounds__(256) void dwpw_kernel(
    const float* __restrict__ QKV,  // B x 768 x 4096
    const float* __restrict__ DW,   // 768 x 25
    const float* __restrict__ PW,   // 768 x 8
    float* __restrict__ AGG)        // B x 768 x 4096
{
    const int p = blockIdx.x * blockDim.x + threadIdx.x;  // 0..4095
    const int g = blockIdx.y;                             // 0..95
    const int b = blockIdx.z;
    const int y = p >> 6, x = p & 63;

    const float* src = QKV + ((size_t)b * C3 + g * 8) * HW;
    float t[8];
    #pragma unroll
    for (int ci = 0; ci < 8; ++ci) {
        const float* ch = src + (size_t)ci * HW;
        const float* wk = DW + (size_t)(g * 8 + ci) * 25;
        float s = 0.f;
        #pragma unroll
        for (int ky = 0; ky < 5; ++ky) {
            const int yy = y + ky - 2;
            if (yy < 0 || yy >= 64) continue;
            #pragma unroll
            for (int kx = 0; kx < 5; ++kx) {
                const int xx = x + kx - 2;
                if (xx < 0 || xx >= 64) continue;
                s = fmaf(ch[yy * 64 + xx], wk[ky * 5 + kx], s);
            }
        }
        t[ci] = s;
    }
    float* dst = AGG + ((size_t)b * C3 + g * 8) * HW + p;
    #pragma unroll
    for (int co = 0; co < 8; ++co) {
        const float* pw = PW + (size_t)(g * 8 + co) * 8;
        float s = 0.f;
        #pragma unroll
        for (int ci = 0; ci < 8; ++ci) s = fmaf(t[ci], pw[ci], s);
        dst[(size_t)co * HW] = s;
    }
}

// ---------------------------------------------------------------------------
// kv[d][e] = sum_n relu(k[n,d]) * v_pad[n,e]  (e==8 -> sum of k)
// ---------------------------------------------------------------------------
__global__ __launch_bounds__(256) void kv_kernel(
    const float* __restrict__ QKV, const float* __restrict__ AGG,
    float* __restrict__ KV)         // (B*64) x 72, layout d*9+e
{
    const int bh = blockIdx.x;
    const int b = bh >> 6, h = bh & 63;
    const float* src = (h < 32) ? (QKV + ((size_t)b * C3 + h * 24) * HW)
                                : (AGG + ((size_t)b * C3 + (h - 32) * 24) * HW);
    const float* Kc = src + (size_t)8  * HW;
    const float* Vc = src + (size_t)16 * HW;

    float acc[72];
    #pragma unroll
    for (int i = 0; i < 72; ++i) acc[i] = 0.f;

    for (int p = threadIdx.x; p < HW; p += 256) {
        float kx[8], vx[8];
        #pragma unroll
        for (int d = 0; d < 8; ++d) kx[d] = fmaxf(Kc[(size_t)d * HW + p], 0.f);
        #pragma unroll
        for (int e = 0; e < 8; ++e) vx[e] = Vc[(size_t)e * HW + p];
        #pragma unroll
        for (int d = 0; d < 8; ++d) {
            #pragma unroll
            for (int e = 0; e < 8; ++e) acc[d * 9 + e] = fmaf(kx[d], vx[e], acc[d * 9 + e]);
            acc[d * 9 + 8] += kx[d];
        }
    }

    __shared__ float red[72];
    for (int i = threadIdx.x; i < 72; i += 256) red[i] = 0.f;
    __syncthreads();
    #pragma unroll
    for (int i = 0; i < 72; ++i) {
        float v = acc[i];
        #pragma unroll
        for (int off = 16; off > 0; off >>= 1) v += __shfl_down(v, off, 32);
        if ((threadIdx.x & 31) == 0) atomicAdd(&red[i], v);
    }
    __syncthreads();
    for (int i = threadIdx.x; i < 72; i += 256) KV[(size_t)bh * 72 + i] = red[i];
}

// ---------------------------------------------------------------------------
// att[n,e] = (sum_d relu(q[n,d]) * kv[d,e]) / (sum_d q kv[d,8] + eps)
// ---------------------------------------------------------------------------
__global__ __launch_bounds__(256) void att_kernel(
    const float* __restrict__ QKV, const float* __restrict__ AGG,
    const float* __restrict__ KV, float* __restrict__ ATT)  // B x 512 x 4096
{
    const int p = blockIdx.x * 256 + threadIdx.x;
    const int h = blockIdx.y, b = blockIdx.z;

    __shared__ float kv[72];
    if (threadIdx.x < 72) kv[threadIdx.x] = KV[(size_t)(b * 64 + h) * 72 + threadIdx.x];
    __syncthreads();

    const float* src = (h < 32) ? (QKV + ((size_t)b * C3 + h * 24) * HW)
                                : (AGG + ((size_t)b * C3 + (h - 32) * 24) * HW);
    float q[8];
    #pragma unroll
    for (int d = 0; d < 8; ++d) q[d] = fmaxf(src[(size_t)d * HW + p], 0.f);

    float o[9];
    #pragma unroll
    for (int e = 0; e < 9; ++e) {
        float s = 0.f;
        #pragma unroll
        for (int d = 0; d < 8; ++d) s = fmaf(q[d], kv[d * 9 + e], s);
        o[e] = s;
    }
    const float invden = 1.0f / (o[8] + 1e-15f);
    float* dst = ATT + ((size_t)b * ATT_C + h * 8) * HW + p;
    #pragma unroll
    for (int e = 0; e < 8; ++e) dst[(size_t)e * HW] = o[e] * invden;
}

// ---------------------------------------------------------------------------
// proj GEMM (M=256, K=512) + batchnorm affine + residual.
// ---------------------------------------------------------------------------
__global__ __launch_bounds__(256) void gemm_proj_kernel(
    const float* __restrict__ Wp,    // 256 x 512
    const float* __restrict__ ATT,   // B x 512 x 4096
    const float* __restrict__ X,     // B x 256 x 4096 (residual)
    const float* __restrict__ gamma, const float* __restrict__ beta,
    const float* __restrict__ mean,  const float* __restrict__ var,
    float* __restrict__ OUT)         // B x 256 x 4096
{
    __shared__ float Bs[16 * BSTRIDE];
    const int tid  = threadIdx.x;
    const int lane = tid & 31;
    const int wave = tid >> 5;
    const int l    = lane & 15;
    const int hi   = lane >> 4;
    const int n0   = blockIdx.x * 64;
    const int m0   = (blockIdx.y * 8 + wave) * 16;
    const int b    = blockIdx.z;

    const float* Ab = ATT + (size_t)b * ATT_C * HW;

    v8f acc0 = {}, acc1 = {}, acc2 = {}, acc3 = {};
    gemm_core<ATT_C>(Wp, Ab, Bs, m0, n0, l, hi, tid, acc0, acc1, acc2, acc3);

    const float* xb = X   + (size_t)b * CIN * HW;
    float*       yb = OUT + (size_t)b * CIN * HW;
    #pragma unroll
    for (int r = 0; r < 8; ++r) {
        const int co = m0 + r + 8 * hi;
        const float inv = gamma[co] * rsqrtf(var[co] + 1e-6f);
        const float bb  = fmaf(-mean[co], inv, beta[co]);
        const size_t base = (size_t)co * HW + n0 + l;
        yb[base + 0]  = fmaf(acc0[r], inv, bb) + xb[base + 0];
        yb[base + 16] = fmaf(acc1[r], inv, bb) + xb[base + 16];
        yb[base + 32] = fmaf(acc2[r], inv, bb) + xb[base + 32];
        yb[base + 48] = fmaf(acc3[r], inv, bb) + xb[base + 48];
    }
}

// ---------------------------------------------------------------------------
extern "C" void kernel_launch(void* const* d_in, const int* in_sizes, int n_in,
                              void* d_out, int out_size, void* d_ws, size_t ws_size,
                              hipStream_t stream) {
    const float* x      = (const float*)d_in[0];
    const float* qkv_w  = (const float*)d_in[1];
    const float* dw_w   = (const float*)d_in[2];
    const float* pw_w   = (const float*)d_in[3];
    const float* proj_w = (const float*)d_in[4];
    const float* gamma  = (const float*)d_in[5];
    const float* beta   = (const float*)d_in[6];
    const float* mean   = (const float*)d_in[7];
    const float* var    = (const float*)d_in[8];
    float* out = (float*)d_out;

    float* qkv = (float*)d_ws;                              // 8*768*4096  (96 MB)
    float* agg = qkv + (size_t)BATCH * C3 * HW;             // 8*768*4096  (96 MB)
    float* att = agg + (size_t)BATCH * C3 * HW;             // 8*512*4096  (64 MB)
    float* kv  = att + (size_t)BATCH * ATT_C * HW;          // 8*64*72

    gemm_qkv_kernel <<<dim3(HW / 64, C3 / 128, BATCH), 256, 0, stream>>>(qkv_w, x, qkv);
    dwpw_kernel     <<<dim3(HW / 256, 96, BATCH),       256, 0, stream>>>(qkv, dw_w, pw_w, agg);
    kv_kernel       <<<dim3(BATCH * 64),                256, 0, stream>>>(qkv, agg, kv);
    att_kernel      <<<dim3(HW / 256, 64, BATCH),       256, 0, stream>>>(qkv, agg, kv, att);
    gemm_proj_kernel<<<dim3(HW / 64, CIN / 128, BATCH), 256, 0, stream>>>(proj_w, att, x, gamma,
                                                                          beta, mean, var, out);
}